// GraphActorCritic_25185688224415
// MI455X (gfx1250) — compile-verified
//
#include <hip/hip_runtime.h>
#include <math.h>

#define NN 50000
#define NE 800000
#define HID 256

typedef float v2f __attribute__((ext_vector_type(2)));
typedef float v8f __attribute__((ext_vector_type(8)));

// ---------------- utility ----------------
__global__ void fill_kernel(float* __restrict__ p, float v, int n) {
  int i = blockIdx.x * blockDim.x + threadIdx.x;
  if (i < n) p[i] = v;
}

// ---------------- edge weight MLP + degree accumulation ----------------
__global__ void __launch_bounds__(256)
edge_mlp_kernel(const int* __restrict__ eids, const int* __restrict__ epos,
                const int* __restrict__ ei,
                const float* __restrict__ id_embed, const float* __restrict__ pos_embed,
                const float* __restrict__ w1, const float* __restrict__ b1,
                const float* __restrict__ w2, const float* __restrict__ b2,
                float* __restrict__ ew_out, float* __restrict__ deg) {
  __shared__ float sW1[100], sB1[10], sW2[10], sB2;
  int t = threadIdx.x;
  if (t < 100) sW1[t] = w1[t];
  if (t < 10) { sB1[t] = b1[t]; sW2[t] = w2[t]; }
  if (t == 0) sB2 = b2[0];
  __syncthreads();
  int e = blockIdx.x * blockDim.x + t;
  if (e >= NE) return;
  float f[10];
  const float* idr = id_embed + (size_t)eids[e] * 8;
#pragma unroll
  for (int i = 0; i < 8; ++i) f[i] = idr[i];
  const float* pr = pos_embed + (size_t)epos[e] * 2;
  f[8] = pr[0]; f[9] = pr[1];
  float z = sB2;
#pragma unroll
  for (int j = 0; j < 10; ++j) {
    float h = sB1[j];
#pragma unroll
    for (int i = 0; i < 10; ++i) h += f[i] * sW1[i * 10 + j];
    h = fmaxf(h, 0.0f);
    z += h * sW2[j];
  }
  float ew = 1.0f / (1.0f + __expf(-z));
  ew_out[e] = ew;
  int dst = ei[NE + e];
  __hip_atomic_fetch_add(&deg[dst], ew, __ATOMIC_RELAXED, __HIP_MEMORY_SCOPE_AGENT);
}

__global__ void rsqrt_kernel(float* __restrict__ p, int n) {
  int i = blockIdx.x * blockDim.x + threadIdx.x;
  if (i < n) p[i] = rsqrtf(p[i]);
}

__global__ void norm_kernel(float* __restrict__ nrm, const int* __restrict__ ei,
                            const float* __restrict__ dinv) {
  int e = blockIdx.x * blockDim.x + threadIdx.x;
  if (e >= NE) return;
  nrm[e] *= dinv[ei[e]] * dinv[ei[NE + e]];
}

// ---------------- conv1: [N,3] @ [3,256] ----------------
__global__ void __launch_bounds__(256)
conv1_kernel(const float* __restrict__ nf, const float* __restrict__ w,
             float* __restrict__ out) {
  int gid = blockIdx.x * blockDim.x + threadIdx.x;
  int n = gid >> 6, c = (gid & 63) << 2;
  if (n >= NN) return;
  float f0 = nf[n * 3 + 0], f1 = nf[n * 3 + 1], f2 = nf[n * 3 + 2];
  float4 w0 = *(const float4*)(w + c);
  float4 w1 = *(const float4*)(w + 256 + c);
  float4 w2 = *(const float4*)(w + 512 + c);
  float4 o;
  o.x = f0 * w0.x + f1 * w1.x + f2 * w2.x;
  o.y = f0 * w0.y + f1 * w1.y + f2 * w2.y;
  o.z = f0 * w0.z + f1 * w1.z + f2 * w2.z;
  o.w = f0 * w0.w + f1 * w1.w + f2 * w2.w;
  *(float4*)(out + (size_t)n * HID + c) = o;
}

// ---------------- edge scatter: agg[dst] += norm*xw[src], wave per edge ----------------
__global__ void __launch_bounds__(256)
scatter_kernel(const float* __restrict__ xw, const float* __restrict__ nrm,
               const int* __restrict__ ei, float* __restrict__ agg) {
  long gid = (long)blockIdx.x * blockDim.x + threadIdx.x;
  int e = (int)(gid >> 5);
  if (e >= NE) return;
  int lane = (int)(gid & 31);
  int src = ei[e], dst = ei[NE + e];
  float s = nrm[e];
  const float4* sp = (const float4*)(xw + (size_t)src * HID) + lane * 2;
  float4 v0 = sp[0], v1 = sp[1];
  float* d = agg + (size_t)dst * HID + lane * 8;
  __hip_atomic_fetch_add(d + 0, s * v0.x, __ATOMIC_RELAXED, __HIP_MEMORY_SCOPE_AGENT);
  __hip_atomic_fetch_add(d + 1, s * v0.y, __ATOMIC_RELAXED, __HIP_MEMORY_SCOPE_AGENT);
  __hip_atomic_fetch_add(d + 2, s * v0.z, __ATOMIC_RELAXED, __HIP_MEMORY_SCOPE_AGENT);
  __hip_atomic_fetch_add(d + 3, s * v0.w, __ATOMIC_RELAXED, __HIP_MEMORY_SCOPE_AGENT);
  __hip_atomic_fetch_add(d + 4, s * v1.x, __ATOMIC_RELAXED, __HIP_MEMORY_SCOPE_AGENT);
  __hip_atomic_fetch_add(d + 5, s * v1.y, __ATOMIC_RELAXED, __HIP_MEMORY_SCOPE_AGENT);
  __hip_atomic_fetch_add(d + 6, s * v1.z, __ATOMIC_RELAXED, __HIP_MEMORY_SCOPE_AGENT);
  __hip_atomic_fetch_add(d + 7, s * v1.w, __ATOMIC_RELAXED, __HIP_MEMORY_SCOPE_AGENT);
}

// ---------------- x = relu(agg + dinv^2*xw + bias) ----------------
__global__ void __launch_bounds__(256)
combine_kernel(const float* __restrict__ agg, const float* __restrict__ xw,
               const float* __restrict__ dinv, const float* __restrict__ bias,
               float* __restrict__ out) {
  int gid = blockIdx.x * blockDim.x + threadIdx.x;
  int n = gid >> 6, c = (gid & 63) << 2;
  if (n >= NN) return;
  float di = dinv[n], ss = di * di;
  float4 a = *(const float4*)(agg + (size_t)n * HID + c);
  float4 x = *(const float4*)(xw + (size_t)n * HID + c);
  float4 b = *(const float4*)(bias + c);
  float4 o;
  o.x = fmaxf(a.x + ss * x.x + b.x, 0.0f);
  o.y = fmaxf(a.y + ss * x.y + b.y, 0.0f);
  o.z = fmaxf(a.z + ss * x.z + b.z, 0.0f);
  o.w = fmaxf(a.w + ss * x.w + b.w, 0.0f);
  *(float4*)(out + (size_t)n * HID + c) = o;
}

// ---------------- f32 WMMA GEMM: C[M,256] = A[M,256] @ W[256,256] (+extra, relu) ----------------
// Block = 8 waves; blockIdx.x = 16-col panel (W panel staged in LDS), blockIdx.y = 128-row tile;
// each wave computes one 16x16 C tile with 64 x V_WMMA_F32_16X16X4_F32.
// HAS_EXTRA/RELU are compile-time so the epilogue is branch-free; the per-lane
// extra[col] load is hoisted ahead of the K-loop (hidden under WMMA latency).
template <bool HAS_EXTRA, bool RELU>
__global__ void __launch_bounds__(256)
gemm_wmma_f32(const float* __restrict__ A, const float* __restrict__ W,
              const float* __restrict__ extra, float* __restrict__ Cout, int M) {
  __shared__ float lw[256 * 16];
  const int n0 = blockIdx.x * 16;
  const int t = threadIdx.x;
  // stage W[:, n0:n0+16] into LDS (16KB)
#pragma unroll
  for (int i = 0; i < 16; ++i) {
    int row = i * 16 + (t >> 4);
    int col = t & 15;
    lw[row * 16 + col] = W[row * 256 + n0 + col];
  }
  __syncthreads();
  const int wave = t >> 5;
  const int lane = t & 31;
  const int mbase = blockIdx.y * 128 + wave * 16;
  if (mbase >= M) return;                 // uniform per wave -> EXEC stays all-1s
  const int half = lane >> 4;             // K-half select for A/B fragments
  const int l16 = lane & 15;              // A: M index ; B/C/D: N index
  const float* arow = A + (size_t)(mbase + l16) * HID;
  float ebias = 0.0f;
  if (HAS_EXTRA) ebias = extra[n0 + l16]; // one load per lane, covered by K-loop
  // prefetch the A row the next y-block will need (same col panel, +128 rows)
  if (mbase + 128 < M) __builtin_prefetch(A + (size_t)(mbase + 128 + l16) * HID, 0, 0);
  v8f acc = {};
  for (int k0 = 0; k0 < 256; k0 += 4) {
    int kk = k0 + 2 * half;
    v2f a, b;
    a.x = arow[kk];
    a.y = arow[kk + 1];                   // contiguous -> global_load_b64
    b.x = lw[kk * 16 + l16];
    b.y = lw[(kk + 1) * 16 + l16];
    acc = __builtin_amdgcn_wmma_f32_16x16x4_f32(
        false, a, false, b, (short)0, acc, false, false);
  }
  // C/D layout: VGPR v holds (M = v + 8*half, N = l16)
#pragma unroll
  for (int v = 0; v < 8; ++v) {
    int row = mbase + v + 8 * half;
    float val = acc[v];
    if (HAS_EXTRA) val += ebias;
    if (RELU) val = fmaxf(val, 0.0f);
    Cout[(size_t)row * HID + n0 + l16] = val;
  }
}

// ---------------- column sums for global mean pooling ----------------
__global__ void __launch_bounds__(256)
mean_kernel(const float* __restrict__ x, float* __restrict__ gsum) {
  int t = threadIdx.x;
  int r0 = blockIdx.x * 256;
  int r1 = r0 + 256; if (r1 > NN) r1 = NN;
  float acc = 0.0f;
  for (int r = r0; r < r1; ++r) acc += x[(size_t)r * HID + t];
  __hip_atomic_fetch_add(&gsum[t], acc, __ATOMIC_RELAXED, __HIP_MEMORY_SCOPE_AGENT);
}

// gterm[j] = actor_b1[j] + sum_k g[k]*actor_w1[256+k, j]
__global__ void gterm_kernel(const float* __restrict__ gsum, const float* __restrict__ aw1,
                             const float* __restrict__ ab1, float* __restrict__ gterm) {
  int j = threadIdx.x;
  const float invN = 1.0f / (float)NN;
  float acc = ab1[j];
  for (int k = 0; k < 256; ++k)
    acc += gsum[k] * invN * aw1[(size_t)(256 + k) * 256 + j];
  gterm[j] = acc;
}

// ---------------- heads ----------------
__global__ void __launch_bounds__(256)
logits_kernel(const float* __restrict__ a, const float* __restrict__ w2,
              const float* __restrict__ b2, float* __restrict__ out) {
  int n = blockIdx.x * blockDim.x + threadIdx.x;
  if (n >= NN) return;
  const float4* ar = (const float4*)(a + (size_t)n * HID);
  const float4* wr = (const float4*)w2;
  float acc = 0.0f;
#pragma unroll 4
  for (int i = 0; i < 64; ++i) {
    float4 av = ar[i], wv = wr[i];
    acc += av.x * wv.x + av.y * wv.y + av.z * wv.z + av.w * wv.w;
  }
  out[n] = acc + b2[0];
}

// pooled = [g, g]; c1[j] = relu(b1[j] + sum_k g[k]*(W1[k,j] + W1[256+k,j]))
__global__ void critic1_kernel(const float* __restrict__ gsum, const float* __restrict__ w1,
                               const float* __restrict__ b1, float* __restrict__ c1) {
  int j = blockIdx.x * blockDim.x + threadIdx.x;
  if (j >= 512) return;
  const float invN = 1.0f / (float)NN;
  float acc = b1[j];
  for (int k = 0; k < 256; ++k) {
    float g = gsum[k] * invN;
    acc += g * (w1[(size_t)k * 512 + j] + w1[(size_t)(256 + k) * 512 + j]);
  }
  c1[j] = fmaxf(acc, 0.0f);
}

__global__ void critic2_kernel(const float* __restrict__ c1, const float* __restrict__ w2,
                               const float* __restrict__ b2, float* __restrict__ c2) {
  int j = threadIdx.x;
  float acc = b2[j];
  for (int k = 0; k < 512; ++k) acc += c1[k] * w2[(size_t)k * 256 + j];
  c2[j] = fmaxf(acc, 0.0f);
}

__global__ void critic3_kernel(const float* __restrict__ c2, const float* __restrict__ w3,
                               const float* __restrict__ b3, float* __restrict__ out) {
  __shared__ float red[256];
  int t = threadIdx.x;
  red[t] = c2[t] * w3[t];
  __syncthreads();
  for (int s = 128; s > 0; s >>= 1) {
    if (t < s) red[t] += red[t + s];
    __syncthreads();
  }
  if (t == 0) out[0] = red[0] + b3[0];
}

// ---------------- orchestration ----------------
extern "C" void kernel_launch(void* const* d_in, const int* in_sizes, int n_in,
                              void* d_out, int out_size, void* d_ws, size_t ws_size,
                              hipStream_t stream) {
  const float* nf       = (const float*)d_in[0];
  const int*   ei       = (const int*)d_in[1];
  const int*   eids     = (const int*)d_in[2];
  const int*   epos     = (const int*)d_in[3];
  const float* id_emb   = (const float*)d_in[4];
  const float* pos_emb  = (const float*)d_in[5];
  const float* ew_w1    = (const float*)d_in[6];
  const float* ew_b1    = (const float*)d_in[7];
  const float* ew_w2    = (const float*)d_in[8];
  const float* ew_b2    = (const float*)d_in[9];
  const float* conv1_w  = (const float*)d_in[10];
  const float* conv1_b  = (const float*)d_in[11];
  const float* conv2_w  = (const float*)d_in[12];
  const float* conv2_b  = (const float*)d_in[13];
  const float* actor_w1 = (const float*)d_in[14];
  const float* actor_b1 = (const float*)d_in[15];
  const float* actor_w2 = (const float*)d_in[16];
  const float* actor_b2 = (const float*)d_in[17];
  const float* crit_w1  = (const float*)d_in[18];
  const float* crit_b1  = (const float*)d_in[19];
  const float* crit_w2  = (const float*)d_in[20];
  const float* crit_b2  = (const float*)d_in[21];
  const float* crit_w3  = (const float*)d_in[22];
  const float* crit_b3  = (const float*)d_in[23];
  float* out = (float*)d_out;

  float* ws   = (float*)d_ws;
  float* nrm  = ws;                       // NE
  float* dinv = nrm + NE;                 // NN   (deg -> dinv in place)
  float* P1   = dinv + NN;                // NN*HID  (xw1, then xw2, then free)
  float* P2   = P1 + (size_t)NN * HID;    // NN*HID  (agg1, agg2, then actor hidden)
  float* P3   = P2 + (size_t)NN * HID;    // NN*HID  (x1, then x2)
  float* gsum = P3 + (size_t)NN * HID;    // 256
  float* gtrm = gsum + 256;               // 256
  float* c1b  = gtrm + 256;               // 512
  float* c2b  = c1b + 512;                // 256

  const int eb = (NE + 255) / 256;                     // 3125
  const int nb64 = (NN * 64 + 255) / 256;              // 12500
  const dim3 ggrid(16, (NN + 127) / 128);              // (16, 391)

  // ---- edge weights + degrees ----
  fill_kernel<<<(NN + 255) / 256, 256, 0, stream>>>(dinv, 1.0f, NN);           // deg = 1 (self loop)
  fill_kernel<<<((int)((size_t)NN * HID) + 255) / 256, 256, 0, stream>>>(P2, 0.0f, NN * HID);
  fill_kernel<<<1, 256, 0, stream>>>(gsum, 0.0f, 256);
  edge_mlp_kernel<<<eb, 256, 0, stream>>>(eids, epos, ei, id_emb, pos_emb,
                                          ew_w1, ew_b1, ew_w2, ew_b2, nrm, dinv);
  rsqrt_kernel<<<(NN + 255) / 256, 256, 0, stream>>>(dinv, NN);
  norm_kernel<<<eb, 256, 0, stream>>>(nrm, ei, dinv);

  // ---- GCN layer 1 ----
  conv1_kernel<<<nb64, 256, 0, stream>>>(nf, conv1_w, P1);
  scatter_kernel<<<(NE * 32) / 256, 256, 0, stream>>>(P1, nrm, ei, P2);
  combine_kernel<<<nb64, 256, 0, stream>>>(P2, P1, dinv, conv1_b, P3);         // x1

  // ---- GCN layer 2 (WMMA GEMM) ----
  gemm_wmma_f32<false, false><<<ggrid, 256, 0, stream>>>(P3, conv2_w, nullptr, P1, NN);  // xw2
  fill_kernel<<<((int)((size_t)NN * HID) + 255) / 256, 256, 0, stream>>>(P2, 0.0f, NN * HID);
  scatter_kernel<<<(NE * 32) / 256, 256, 0, stream>>>(P1, nrm, ei, P2);
  combine_kernel<<<nb64, 256, 0, stream>>>(P2, P1, dinv, conv2_b, P3);         // x2

  // ---- pooling + heads ----
  mean_kernel<<<(NN + 255) / 256, 256, 0, stream>>>(P3, gsum);
  gterm_kernel<<<1, 256, 0, stream>>>(gsum, actor_w1, actor_b1, gtrm);
  gemm_wmma_f32<true, true><<<ggrid, 256, 0, stream>>>(P3, actor_w1, gtrm, P2, NN);      // actor hidden
  logits_kernel<<<(NN + 255) / 256, 256, 0, stream>>>(P2, actor_w2, actor_b2, out);
  critic1_kernel<<<2, 256, 0, stream>>>(gsum, crit_w1, crit_b1, c1b);
  critic2_kernel<<<1, 256, 0, stream>>>(c1b, crit_w2, crit_b2, c2b);
  critic3_kernel<<<1, 256, 0, stream>>>(c2b, crit_w3, crit_b3, out + NN);
  (void)in_sizes; (void)n_in; (void)out_size; (void)ws_size;
}